// KGIN_52441550684533
// MI455X (gfx1250) — compile-verified
//
#include <hip/hip_runtime.h>
#include <hip/hip_bf16.h>
#include <math.h>

#define DIM 64

typedef float v2f __attribute__((ext_vector_type(2)));
typedef float v8f __attribute__((ext_vector_type(8)));

// ---------------------------------------------------------------------------
// Prep: gate = sigmoid(relation_emb) [32x64]; intent_emb = softmax(intent_w)@relation_emb [2x64]
// ---------------------------------------------------------------------------
__global__ __launch_bounds__(256) void kgin_prep(const float* __restrict__ rel_emb,
                                                 const float* __restrict__ iw,
                                                 float* __restrict__ gate,
                                                 float* __restrict__ intent_emb) {
  int t = threadIdx.x;
  for (int idx = t; idx < 32 * DIM; idx += 256)
    gate[idx] = 1.0f / (1.0f + expf(-rel_emb[idx]));
  if (t < DIM) {
    for (int j = 0; j < 2; ++j) {
      float mx = -3.402823466e+38f;
      for (int r = 0; r < 32; ++r) mx = fmaxf(mx, iw[j * 32 + r]);
      float den = 0.f, acc = 0.f;
      for (int r = 0; r < 32; ++r) {
        float e = expf(iw[j * 32 + r] - mx);
        den += e;
        acc += e * rel_emb[r * DIM + t];
      }
      intent_emb[j * DIM + t] = acc / den;
    }
  }
}

// ---------------------------------------------------------------------------
// Y[m][n] = sum_k E[m][k] * W[n][k]   (M x 64 @ 64 x 64, f32 WMMA 16x16x4)
// 128 threads = 4 waves; each wave computes a 16x64 output tile.
// ---------------------------------------------------------------------------
__global__ __launch_bounds__(128) void kgin_gemm(const float* __restrict__ E,
                                                 const float* __restrict__ W,
                                                 float* __restrict__ Y, int M) {
  __shared__ float sW[64][65];
  __shared__ float sA[64][65];
  const int tid  = threadIdx.x;
  const int wave = tid >> 5;
  const int lane = tid & 31;
  const int m0   = blockIdx.x * 64;

  for (int idx = tid; idx < 64 * 64; idx += 128) {
    int r = idx >> 6, c = idx & 63;
    sW[r][c] = W[idx];
    int row = m0 + r;
    sA[r][c] = (row < M) ? E[(size_t)row * DIM + c] : 0.f;
  }
  __syncthreads();

  const int h = lane >> 4;   // lane half selects K sub-pair
  const int l = lane & 15;   // row (A) / col (B,D) within tile
  const int mrow = wave * 16 + l;

  v8f acc[4] = {};
#pragma unroll
  for (int kk = 0; kk < 16; ++kk) {
    const int kb = kk * 4 + 2 * h;
    v2f a;
    a.x = sA[mrow][kb];
    a.y = sA[mrow][kb + 1];
#pragma unroll
    for (int nt = 0; nt < 4; ++nt) {
      const int n = nt * 16 + l;
      v2f b;
      b.x = sW[n][kb];
      b.y = sW[n][kb + 1];
      acc[nt] = __builtin_amdgcn_wmma_f32_16x16x4_f32(
          /*neg_a=*/false, a, /*neg_b=*/false, b,
          /*c_mod=*/(short)0, acc[nt], /*reuse_a=*/false, /*reuse_b=*/false);
    }
  }

#pragma unroll
  for (int nt = 0; nt < 4; ++nt)
#pragma unroll
    for (int j = 0; j < 8; ++j) {
      int row = m0 + wave * 16 + j + 8 * h;
      if (row < M) Y[(size_t)row * DIM + nt * 16 + l] = acc[nt][j];
    }
}

// ---------------------------------------------------------------------------
// KG edge scatter: agg[dst] += Y[src] * gate[rel]; deg[dst] += 1.  Warp per edge.
// ---------------------------------------------------------------------------
__global__ __launch_bounds__(256) void kgin_kg_scatter(const int* __restrict__ src,
                                                       const int* __restrict__ dst,
                                                       const int* __restrict__ rel,
                                                       const float* __restrict__ Y,
                                                       const float* __restrict__ gate,
                                                       float* __restrict__ agg,
                                                       float* __restrict__ deg, int nE) {
  int e = blockIdx.x * 8 + (threadIdx.x >> 5);
  if (e >= nE) return;
  int lane = threadIdx.x & 31;
  int s = src[e], d = dst[e], r = rel[e];
  float2 v = *(const float2*)(Y + (size_t)s * DIM + lane * 2);
  float2 g = *(const float2*)(gate + (size_t)r * DIM + lane * 2);
  float* ad = agg + (size_t)d * DIM + lane * 2;
  atomicAdd(ad, v.x * g.x);
  atomicAdd(ad + 1, v.y * g.y);
  if (lane == 0) atomicAdd(deg + d, 1.0f);
}

// ---------------------------------------------------------------------------
// X = l2norm(elu(agg/deg + E)).  Warp per entity row.
// ---------------------------------------------------------------------------
__global__ __launch_bounds__(256) void kgin_kg_update(const float* __restrict__ agg,
                                                      const float* __restrict__ deg,
                                                      const float* __restrict__ E,
                                                      float* __restrict__ X, int n) {
  int m = blockIdx.x * 8 + (threadIdx.x >> 5);
  if (m >= n) return;
  int lane = threadIdx.x & 31;
  float dg = fmaxf(deg[m], 1.0f);
  size_t base = (size_t)m * DIM;
  float a0 = agg[base + lane] / dg + E[base + lane];
  float a1 = agg[base + lane + 32] / dg + E[base + lane + 32];
  a0 = a0 > 0.f ? a0 : expm1f(a0);
  a1 = a1 > 0.f ? a1 : expm1f(a1);
  float ss = a0 * a0 + a1 * a1;
  for (int off = 16; off; off >>= 1) ss += __shfl_xor(ss, off, 32);
  float inv = 1.0f / fmaxf(sqrtf(ss), 1e-12f);
  X[base + lane] = a0 * inv;
  X[base + lane + 32] = a1 * inv;
}

// ---------------------------------------------------------------------------
// item_kg[it] = X[item2entity[it]].  Warp per item.
// ---------------------------------------------------------------------------
__global__ __launch_bounds__(256) void kgin_gather_items(const float* __restrict__ X,
                                                         const int* __restrict__ i2e,
                                                         float* __restrict__ IK, int nItems) {
  int it = blockIdx.x * 8 + (threadIdx.x >> 5);
  if (it >= nItems) return;
  int lane = threadIdx.x & 31;
  size_t sb = (size_t)i2e[it] * DIM;
  float2 v = *(const float2*)(X + sb + lane * 2);
  *(float2*)(IK + (size_t)it * DIM + lane * 2) = v;
}

// ---------------------------------------------------------------------------
// UI scatter: agg[row] += val * all_emb[col] (users -> user_emb, items -> item_kg)
// ---------------------------------------------------------------------------
__global__ __launch_bounds__(256) void kgin_ui_scatter(const int* __restrict__ row,
                                                       const int* __restrict__ col,
                                                       const float* __restrict__ vals,
                                                       const float* __restrict__ UE,
                                                       const float* __restrict__ IK,
                                                       float* __restrict__ agg,
                                                       int nE, int nUsers) {
  int e = blockIdx.x * 8 + (threadIdx.x >> 5);
  if (e >= nE) return;
  int lane = threadIdx.x & 31;
  int r = row[e], c = col[e];
  float v = vals[e];
  const float* srcp = (c < nUsers) ? (UE + (size_t)c * DIM)
                                   : (IK + (size_t)(c - nUsers) * DIM);
  float2 x = *(const float2*)(srcp + lane * 2);
  float* dp = agg + (size_t)r * DIM + lane * 2;
  atomicAdd(dp, v * x.x);
  atomicAdd(dp + 1, v * x.y);
}

// ---------------------------------------------------------------------------
// Final scoring: warp per batch element; router softmax (2 intents) + dots.
// ---------------------------------------------------------------------------
__global__ __launch_bounds__(128) void kgin_score(const int* __restrict__ u,
                                                  const int* __restrict__ ipos,
                                                  const int* __restrict__ ineg,
                                                  const float* __restrict__ UE,
                                                  const float* __restrict__ IK,
                                                  const float* __restrict__ agg,
                                                  const float* __restrict__ rw,
                                                  const float* __restrict__ rb,
                                                  const float* __restrict__ ieb,
                                                  float* __restrict__ out,
                                                  int B, int nUsers) {
  int b = blockIdx.x * 4 + (threadIdx.x >> 5);
  if (b >= B) return;
  int lane = threadIdx.x & 31;
  int uu = u[b];
  size_t ub = (size_t)uu * DIM;
  float up0 = 0.5f * (UE[ub + lane] + agg[ub + lane]);
  float up1 = 0.5f * (UE[ub + lane + 32] + agg[ub + lane + 32]);
  float l0 = up0 * rw[lane] + up1 * rw[lane + 32];
  float l1 = up0 * rw[DIM + lane] + up1 * rw[DIM + lane + 32];
  for (int off = 16; off; off >>= 1) {
    l0 += __shfl_xor(l0, off, 32);
    l1 += __shfl_xor(l1, off, 32);
  }
  l0 += rb[0];
  l1 += rb[1];
  float mx = fmaxf(l0, l1);
  float e0 = expf(l0 - mx), e1 = expf(l1 - mx);
  float inv = 1.0f / (e0 + e1);
  float p0 = e0 * inv, p1 = e1 * inv;
  float ue0 = up0 + p0 * ieb[lane] + p1 * ieb[DIM + lane];
  float ue1 = up1 + p0 * ieb[lane + 32] + p1 * ieb[DIM + lane + 32];

  for (int which = 0; which < 2; ++which) {
    int it = which == 0 ? ipos[b] : ineg[b];
    size_t node = (size_t)(nUsers + it) * DIM;
    size_t ib = (size_t)it * DIM;
    float ie0 = 1.5f * IK[ib + lane] + 0.5f * agg[node + lane];
    float ie1 = 1.5f * IK[ib + lane + 32] + 0.5f * agg[node + lane + 32];
    float dot = ue0 * ie0 + ue1 * ie1;
    for (int off = 16; off; off >>= 1) dot += __shfl_xor(dot, off, 32);
    if (lane == 0) out[which == 0 ? b : B + b] = dot;
  }
}

// ---------------------------------------------------------------------------
extern "C" void kernel_launch(void* const* d_in, const int* in_sizes, int n_in,
                              void* d_out, int out_size, void* d_ws, size_t ws_size,
                              hipStream_t stream) {
  (void)n_in; (void)out_size; (void)ws_size;
  const int*   u        = (const int*)  d_in[0];
  const int*   ipos     = (const int*)  d_in[1];
  const int*   ineg     = (const int*)  d_in[2];
  const float* user_emb = (const float*)d_in[3];
  const float* ent_emb  = (const float*)d_in[4];
  const float* rel_emb  = (const float*)d_in[5];
  const float* iw       = (const float*)d_in[6];
  const float* rw       = (const float*)d_in[7];
  const float* rb       = (const float*)d_in[8];
  const float* kgw      = (const float*)d_in[9];   // (1,64,64)
  const float* ui_vals  = (const float*)d_in[10];
  const int*   i2e      = (const int*)  d_in[11];
  const int*   kg_src   = (const int*)  d_in[12];
  const int*   kg_dst   = (const int*)  d_in[13];
  const int*   kg_rel   = (const int*)  d_in[14];
  const int*   ui_row   = (const int*)  d_in[15];
  const int*   ui_col   = (const int*)  d_in[16];
  float* out = (float*)d_out;

  const int B      = in_sizes[0];
  const int nUsers = in_sizes[3] / DIM;
  const int nEnt   = in_sizes[4] / DIM;
  const int nItems = in_sizes[11];
  const int eKG    = in_sizes[12];
  const int eUI    = in_sizes[15];

  float* ws = (float*)d_ws;
  size_t off = 0;
  auto alloc = [&](size_t n) {
    float* p = ws + off;
    off = (off + n + 63) & ~(size_t)63;
    return p;
  };
  float* deg  = alloc((size_t)nEnt);
  float* gate = alloc(32 * DIM);
  float* ieb  = alloc(2 * DIM);
  float* Y    = alloc((size_t)nEnt * DIM);   // transformed entities, reused as normalized X
  float* AGG  = alloc((size_t)nEnt * DIM);   // KG aggregation, reused as UI aggregation
  float* IK   = alloc((size_t)nItems * DIM); // item_kg

  hipMemsetAsync(deg, 0, (size_t)nEnt * sizeof(float), stream);
  hipMemsetAsync(AGG, 0, (size_t)nEnt * DIM * sizeof(float), stream);

  kgin_prep<<<1, 256, 0, stream>>>(rel_emb, iw, gate, ieb);
  kgin_gemm<<<(nEnt + 63) / 64, 128, 0, stream>>>(ent_emb, kgw, Y, nEnt);
  kgin_kg_scatter<<<(eKG + 7) / 8, 256, 0, stream>>>(kg_src, kg_dst, kg_rel, Y, gate, AGG, deg, eKG);
  kgin_kg_update<<<(nEnt + 7) / 8, 256, 0, stream>>>(AGG, deg, ent_emb, Y, nEnt);

  hipMemsetAsync(AGG, 0, (size_t)(nUsers + nItems) * DIM * sizeof(float), stream);
  kgin_gather_items<<<(nItems + 7) / 8, 256, 0, stream>>>(Y, i2e, IK, nItems);
  kgin_ui_scatter<<<(eUI + 7) / 8, 256, 0, stream>>>(ui_row, ui_col, ui_vals, user_emb, IK, AGG, eUI, nUsers);
  kgin_score<<<(B + 3) / 4, 128, 0, stream>>>(u, ipos, ineg, user_emb, IK, AGG, rw, rb, ieb, out, B, nUsers);
}